// BCELoss_75411035783735
// MI455X (gfx1250) — compile-verified
//
#include <hip/hip_runtime.h>
#include <hip/hip_bf16.h>
#include <math.h>

#ifndef __has_builtin
#define __has_builtin(x) 0
#endif

#if __has_builtin(__builtin_amdgcn_tensor_load_to_lds)
#define USE_TDM 1
#else
#define USE_TDM 0
#endif

#if __has_builtin(__builtin_amdgcn_wmma_f32_16x16x4_f32)
#define USE_WMMA 1
#else
#define USE_WMMA 0
#endif

#define TILE 2048            // 8 KB per TDM descriptor, 72 KB LDS/WG total
#define NTHREADS 256
#define K1_BLOCKS 400
#define KREF_BLOCKS 1024
#define ELEMS_PER_THREAD (TILE / NTHREADS)   // 8
#define VEC_PER_THREAD   (TILE / 4 / NTHREADS) // 2 float4s

typedef float v2f  __attribute__((ext_vector_type(2)));
typedef float v8f  __attribute__((ext_vector_type(8)));
typedef unsigned int u32x4 __attribute__((ext_vector_type(4)));
typedef int i32x4 __attribute__((ext_vector_type(4)));
typedef int i32x8 __attribute__((ext_vector_type(8)));

// ---------------- workspace layout (zeroed each launch via hipMemsetAsync) ---
struct OhemWS {
  unsigned long long pos_cnt;        // number of positive elements
  unsigned long long pos_sum_fx;     // positive loss sum, fixed point (x 2^32)
  unsigned long long k_sel;          // negative_count actually selected
  unsigned long long cnt_above;      // elems strictly above current threshold
  unsigned long long sum_above_fx;   // their loss sum (fixed point)
  unsigned long long prefix;         // radix prefix of threshold bin
  unsigned long long h0_sum[2048];   // level 0: f32 bits [31:21]
  unsigned long long h1_sum[2048];   // level 1: f32 bits [20:10]
  unsigned long long h2_sum[1024];   // level 2: f32 bits [ 9: 0]
  unsigned int       h0_cnt[2048];
  unsigned int       h1_cnt[2048];
  unsigned int       h2_cnt[1024];
};

// ---------------- helpers ----------------------------------------------------
__device__ __forceinline__ float bce_logits(float x, float t) {
  // max(x,0) - x*t + log1p(exp(-|x|)) ; always >= 0 for t in {0,1}
  return fmaxf(x, 0.f) - x * t + log1pf(expf(-fabsf(x)));
}

__device__ __forceinline__ unsigned long long to_fx(float v) {
  return (unsigned long long)((double)v * 4294967296.0);
}

__device__ __forceinline__ unsigned lds_byte_off(const void* p) {
  // generic pointer to LDS: low 32 bits are the LDS byte offset
  return (unsigned)(size_t)p;
}

#if USE_TDM
// 1-row (1D) TDM tile descriptor: tile_dim0 elements of 4B starting at gsrc.
// tensor_dim0 = elements remaining so OOB reads return zero into LDS.
__device__ __forceinline__ void tdm_load_row(const float* gsrc, unsigned lds_off,
                                             unsigned avail_elems, unsigned tile_elems) {
  unsigned long long ga = (unsigned long long)(size_t)gsrc;
  u32x4 g0;
  g0[0] = 1u;                                                  // count=1, user mode
  g0[1] = lds_off;                                             // lds_addr (bytes)
  g0[2] = (unsigned)ga;                                        // global_addr[31:0]
  g0[3] = ((unsigned)(ga >> 32) & 0x01FFFFFFu) | 0x80000000u;  // addr[56:32] | type=2
  i32x8 g1;
  g1[0] = (int)(2u << 16);                                     // data_size = 4B
  g1[1] = (int)((avail_elems & 0xFFFFu) << 16);                // tensor_dim0 lo16
  g1[2] = (int)((avail_elems >> 16) | (1u << 16));             // dim0 hi | tensor_dim1=1
  g1[3] = (int)(tile_elems << 16);                             // dim1 hi=0 | tile_dim0
  g1[4] = 1;                                                   // tile_dim1=1, tile_dim2=0
  g1[5] = (int)avail_elems;                                    // tensor_dim0_stride lo32
  g1[6] = 0;                                                   // stride hi | dim1 stride
  g1[7] = 0;
  i32x4 z4 = {0, 0, 0, 0};
#if __has_include(<hip/amd_detail/amd_gfx1250_TDM.h>)
  i32x8 z8 = {0, 0, 0, 0, 0, 0, 0, 0};
  __builtin_amdgcn_tensor_load_to_lds(g0, g1, z4, z4, z8, 0);
#else
  __builtin_amdgcn_tensor_load_to_lds(g0, g1, z4, z4, 0);
#endif
}
#endif

// ---------------- pass 1: counts, pos-sum, level-0 histogram ------------------
__global__ __launch_bounds__(NTHREADS) void ohem_k1_main(
    const float* __restrict__ pred, const float* __restrict__ gt,
    const float* __restrict__ mask, OhemWS* __restrict__ ws, int n) {
  __shared__ __align__(16) float sh[2][3][TILE];   // double-buffered staging
  __shared__ unsigned int       shCnt[2048];
  __shared__ unsigned long long shSum[2048];
  __shared__ float              shWaveSum[NTHREADS / 32];
  __shared__ unsigned int       shWaveCnt[NTHREADS / 32];

  const int tid = threadIdx.x;
  for (int i = tid; i < 2048; i += NTHREADS) { shCnt[i] = 0u; shSum[i] = 0ull; }
  __syncthreads();

  const long long ntile  = ((long long)n + TILE - 1) / TILE;
  const long long stride = gridDim.x;
  const long long t0     = blockIdx.x;

  float        posSum = 0.f;
  unsigned int posCnt = 0u;

#if USE_TDM
  auto issue_tile = [&](long long t, int buf) {
    long long base = t * (long long)TILE;
    unsigned avail = (unsigned)((long long)n - base);
    tdm_load_row(pred + base, lds_byte_off(&sh[buf][0][0]), avail, TILE);
    tdm_load_row(gt   + base, lds_byte_off(&sh[buf][1][0]), avail, TILE);
    tdm_load_row(mask + base, lds_byte_off(&sh[buf][2][0]), avail, TILE);
  };
  if (tid == 0 && t0 < ntile) issue_tile(t0, 0);   // prologue prefetch
#endif

  int cur = 0;
  for (long long t = t0; t < ntile; t += stride) {
#if USE_TDM
    if (tid == 0) {
      long long tn = t + stride;
      if (tn < ntile) {
        issue_tile(tn, cur ^ 1);                     // prefetch next tile
        __builtin_amdgcn_s_wait_tensorcnt(3);        // wait for current 3 loads
      } else {
        __builtin_amdgcn_s_wait_tensorcnt(0);
      }
    }
#else
    {
      long long base = t * (long long)TILE;
      for (int i = tid; i < TILE; i += NTHREADS) {
        long long g = base + i;
        bool ok = g < (long long)n;
        sh[cur][0][i] = ok ? pred[g] : 0.f;
        sh[cur][1][i] = ok ? gt[g]   : 0.f;
        sh[cur][2][i] = ok ? mask[g] : 0.f;   // mask=0 => element is inert
      }
    }
#endif
    __syncthreads();

#pragma unroll
    for (int v = 0; v < VEC_PER_THREAD; ++v) {
      const int idx = tid + v * NTHREADS;
      const float4 xp = reinterpret_cast<const float4*>(&sh[cur][0][0])[idx];
      const float4 xg = reinterpret_cast<const float4*>(&sh[cur][1][0])[idx];
      const float4 xm = reinterpret_cast<const float4*>(&sh[cur][2][0])[idx];
      const float px[4] = {xp.x, xp.y, xp.z, xp.w};
      const float gx[4] = {xg.x, xg.y, xg.z, xg.w};
      const float mx[4] = {xm.x, xm.y, xm.z, xm.w};
#pragma unroll
      for (int e = 0; e < 4; ++e) {
        float g = gx[e], m = mx[e];
        float loss = bce_logits(px[e], g);
        if ((g * m) > 0.f)         { posCnt++; posSum += loss; }
        if (((1.f - g) * m) > 0.f) {
          unsigned bits = __float_as_uint(loss);
          unsigned b    = bits >> 21;                // top 11 bits (sign is 0)
          atomicAdd(&shCnt[b], 1u);
          atomicAdd(&shSum[b], to_fx(loss));
        }
      }
    }
    __syncthreads();
    cur ^= 1;
  }

  // flush block-private histogram (integer atomics -> deterministic)
  for (int i = tid; i < 2048; i += NTHREADS) {
    unsigned c = shCnt[i];
    if (c) {
      atomicAdd(&ws->h0_cnt[i], c);
      atomicAdd(&ws->h0_sum[i], shSum[i]);
    }
  }

  // per-wave reduction: counts via shuffles, sums via the matrix pipe
  unsigned pc = posCnt;
  for (int o = 16; o; o >>= 1) pc += __shfl_xor(pc, o, 32);

  float wsum;
#if USE_WMMA
  {
    // D = Ones(16x4) * B(4x16): every D[m][n] is a column sum of B, so the
    // 32 per-lane partials (B = {p, 0}) are reduced by one v_wmma_f32_16x16x4_f32.
    v2f a; a[0] = 1.f;    a[1] = 1.f;
    v2f b; b[0] = posSum; b[1] = 0.f;
    v8f c = {};
    c = __builtin_amdgcn_wmma_f32_16x16x4_f32(false, a, false, b, (short)0, c,
                                              false, false);
    wsum = c[0];                                   // col-sum, duplicated per half
    for (int o = 16; o; o >>= 1) wsum += __shfl_xor(wsum, o, 32);
    wsum *= 0.5f;                                  // each column counted twice
  }
#else
  wsum = posSum;
  for (int o = 16; o; o >>= 1) wsum += __shfl_xor(wsum, o, 32);
#endif

  const int wave = tid >> 5, lane = tid & 31;
  if (lane == 0) { shWaveSum[wave] = wsum; shWaveCnt[wave] = pc; }
  __syncthreads();
  if (tid == 0) {
    float bs = 0.f; unsigned long long bc = 0ull;
    for (int w = 0; w < NTHREADS / 32; ++w) { bs += shWaveSum[w]; bc += shWaveCnt[w]; }
    atomicAdd(&ws->pos_cnt, bc);
    atomicAdd(&ws->pos_sum_fx, to_fx(bs));
  }
}

// ---------------- refinement passes (L2-resident re-reads) -------------------
__global__ __launch_bounds__(NTHREADS) void ohem_k3_level1(
    const float* __restrict__ pred, const float* __restrict__ gt,
    const float* __restrict__ mask, OhemWS* __restrict__ ws, int n) {
  __shared__ unsigned int       shCnt[2048];
  __shared__ unsigned long long shSum[2048];
  for (int i = threadIdx.x; i < 2048; i += NTHREADS) { shCnt[i] = 0u; shSum[i] = 0ull; }
  __syncthreads();

  const unsigned pref = (unsigned)ws->prefix;      // level-0 threshold bin
  const int gid = blockIdx.x * NTHREADS + threadIdx.x;
  const int nth = gridDim.x * NTHREADS;
  const int n4  = n >> 2;

  auto touch = [&](float x, float g, float m) {
    if (((1.f - g) * m) > 0.f) {
      float loss = bce_logits(x, g);
      unsigned bits = __float_as_uint(loss);
      if ((bits >> 21) == pref) {
        unsigned b = (bits >> 10) & 0x7FFu;
        atomicAdd(&shCnt[b], 1u);
        atomicAdd(&shSum[b], to_fx(loss));
      }
    }
  };
  const float4* p4 = reinterpret_cast<const float4*>(pred);
  const float4* g4 = reinterpret_cast<const float4*>(gt);
  const float4* m4 = reinterpret_cast<const float4*>(mask);
  for (int i = gid; i < n4; i += nth) {
    float4 x = p4[i], g = g4[i], m = m4[i];
    touch(x.x, g.x, m.x); touch(x.y, g.y, m.y);
    touch(x.z, g.z, m.z); touch(x.w, g.w, m.w);
  }
  for (int i = 4 * n4 + gid; i < n; i += nth) touch(pred[i], gt[i], mask[i]);

  __syncthreads();
  for (int i = threadIdx.x; i < 2048; i += NTHREADS) {
    unsigned c = shCnt[i];
    if (c) { atomicAdd(&ws->h1_cnt[i], c); atomicAdd(&ws->h1_sum[i], shSum[i]); }
  }
}

__global__ __launch_bounds__(NTHREADS) void ohem_k5_level2(
    const float* __restrict__ pred, const float* __restrict__ gt,
    const float* __restrict__ mask, OhemWS* __restrict__ ws, int n) {
  __shared__ unsigned int       shCnt[1024];
  __shared__ unsigned long long shSum[1024];
  for (int i = threadIdx.x; i < 1024; i += NTHREADS) { shCnt[i] = 0u; shSum[i] = 0ull; }
  __syncthreads();

  const unsigned pref = (unsigned)ws->prefix;      // 22-bit prefix (b0<<11)|b1
  const int gid = blockIdx.x * NTHREADS + threadIdx.x;
  const int nth = gridDim.x * NTHREADS;
  const int n4  = n >> 2;

  auto touch = [&](float x, float g, float m) {
    if (((1.f - g) * m) > 0.f) {
      float loss = bce_logits(x, g);
      unsigned bits = __float_as_uint(loss);
      if ((bits >> 10) == pref) {
        unsigned b = bits & 0x3FFu;
        atomicAdd(&shCnt[b], 1u);
        atomicAdd(&shSum[b], to_fx(loss));
      }
    }
  };
  const float4* p4 = reinterpret_cast<const float4*>(pred);
  const float4* g4 = reinterpret_cast<const float4*>(gt);
  const float4* m4 = reinterpret_cast<const float4*>(mask);
  for (int i = gid; i < n4; i += nth) {
    float4 x = p4[i], g = g4[i], m = m4[i];
    touch(x.x, g.x, m.x); touch(x.y, g.y, m.y);
    touch(x.z, g.z, m.z); touch(x.w, g.w, m.w);
  }
  for (int i = 4 * n4 + gid; i < n; i += nth) touch(pred[i], gt[i], mask[i]);

  __syncthreads();
  for (int i = threadIdx.x; i < 1024; i += NTHREADS) {
    unsigned c = shCnt[i];
    if (c) { atomicAdd(&ws->h2_cnt[i], c); atomicAdd(&ws->h2_sum[i], shSum[i]); }
  }
}

// ---------------- tiny scan kernels (single thread over <=2048 bins) ---------
__global__ void ohem_scan0(OhemWS* ws) {
  if (threadIdx.x != 0 || blockIdx.x != 0) return;
  unsigned long long total = 0ull;
  for (int i = 0; i < 2048; ++i) total += (unsigned long long)ws->h0_cnt[i];
  unsigned long long cap =
      (unsigned long long)floorf((float)ws->pos_cnt * 3.0f);   // NEGATIVE_RATIO
  unsigned long long k = total < cap ? total : cap;
  ws->k_sel = k;
  unsigned long long cum = 0ull, sum = 0ull; int b0 = 2047;
  for (int b = 2047; b >= 0; --b) {
    unsigned long long c = (unsigned long long)ws->h0_cnt[b];
    if (cum + c >= k) { b0 = b; break; }
    cum += c; sum += ws->h0_sum[b];
  }
  ws->cnt_above = cum; ws->sum_above_fx = sum;
  ws->prefix = (unsigned long long)b0;
}

__global__ void ohem_scan1(OhemWS* ws) {
  if (threadIdx.x != 0 || blockIdx.x != 0) return;
  unsigned long long k = ws->k_sel - ws->cnt_above;
  unsigned long long cum = 0ull, sum = 0ull; int b1 = 2047;
  for (int b = 2047; b >= 0; --b) {
    unsigned long long c = (unsigned long long)ws->h1_cnt[b];
    if (cum + c >= k) { b1 = b; break; }
    cum += c; sum += ws->h1_sum[b];
  }
  ws->cnt_above += cum; ws->sum_above_fx += sum;
  ws->prefix = (ws->prefix << 11) | (unsigned long long)b1;
}

__global__ void ohem_scan2(OhemWS* ws, float* out) {
  if (threadIdx.x != 0 || blockIdx.x != 0) return;
  unsigned long long k = ws->k_sel - ws->cnt_above;
  unsigned long long cum = 0ull, sum = 0ull; int b2 = 1023;
  for (int b = 1023; b >= 0; --b) {
    unsigned long long c = (unsigned long long)ws->h2_cnt[b];
    if (cum + c >= k) { b2 = b; break; }
    cum += c; sum += ws->h2_sum[b];
  }
  unsigned long long cnt_ab = ws->cnt_above + cum;
  unsigned long long sum_ab = ws->sum_above_fx + sum;
  unsigned long long r      = ws->k_sel - cnt_ab;     // ties at the k-th value
  unsigned bits = (unsigned)((ws->prefix << 10) | (unsigned long long)b2);
  double v       = (double)__uint_as_float(bits);     // exact k-th loss value
  double neg_top = (double)sum_ab * 0x1p-32 + (double)r * v;
  double pos_sum = (double)ws->pos_sum_fx * 0x1p-32;
  double denom   = (double)ws->pos_cnt + (double)ws->k_sel + 1e-6;
  *out = (float)((pos_sum + neg_top) / denom);
}

// ---------------- host entry --------------------------------------------------
extern "C" void kernel_launch(void* const* d_in, const int* in_sizes, int n_in,
                              void* d_out, int out_size, void* d_ws, size_t ws_size,
                              hipStream_t stream) {
  const float* pred = (const float*)d_in[0];
  const float* gt   = (const float*)d_in[1];
  const float* mask = (const float*)d_in[2];
  OhemWS* ws = (OhemWS*)d_ws;
  const int n = in_sizes[0];

  hipMemsetAsync(d_ws, 0, sizeof(OhemWS), stream);
  ohem_k1_main<<<K1_BLOCKS, NTHREADS, 0, stream>>>(pred, gt, mask, ws, n);
  ohem_scan0<<<1, 32, 0, stream>>>(ws);
  ohem_k3_level1<<<KREF_BLOCKS, NTHREADS, 0, stream>>>(pred, gt, mask, ws, n);
  ohem_scan1<<<1, 32, 0, stream>>>(ws);
  ohem_k5_level2<<<KREF_BLOCKS, NTHREADS, 0, stream>>>(pred, gt, mask, ws, n);
  ohem_scan2<<<1, 32, 0, stream>>>(ws, (float*)d_out);
}